// LogNormCell_80453327389343
// MI455X (gfx1250) — compile-verified
//
#include <hip/hip_runtime.h>
#include <hip/hip_bf16.h>
#include <stdint.h>

// Problem shape (fixed by the reference): inputs [B=32, T=4096, D=256] fp32.
#define B_DIM   32
#define T_DIM   4096
#define D_DIM   256
#define NT      256            // threads per block (= D), 8 wave32 waves
#define CH      128            // T elements per chunk
#define NCHUNK  (T_DIM / CH)   // 32 chunks -> grid 32x32 = 1024 workgroups
#define BUF_T   32             // T rows per async stage
#define NSTAGE  (CH / BUF_T)   // 4 stages per chunk
#define STAGE_F (BUF_T * D_DIM)     // 8192 floats per stage
#define STAGE_BYTES (STAGE_F * 4)   // 32 KB per stage; double-buffered -> 64 KB LDS

// ---------------- CDNA5 async global->LDS pipeline ----------------
// ISA 15.18.3 opcode 98: copies global memory at vaddr into LDS at vdst
// INST_OFFSET applies to BOTH sides, so a mirrored LDS layout needs only one
// base address register per stage and 8 immediate-offset issues.
template <int OFF>
__device__ __forceinline__ void async_copy16(uint32_t lds_byte, const float* gaddr) {
  asm volatile("global_load_async_to_lds_b128 %0, %1, off offset:%2"
               :
               : "v"(lds_byte), "v"(gaddr), "n"(OFF)
               : "memory");
}

#define WAIT_ASYNC(n) asm volatile("s_wait_asynccnt " #n ::: "memory")

// Issue one 32 KB stage: each of 256 lanes loads 8 x 16B (b128), ASYNCcnt += 8/wave.
__device__ __forceinline__ void issue_stage(const float* gstage, uint32_t lds_stage, int tid) {
  const float* ga = gstage + 4 * tid;      // 16 B per lane, coalesced 512 B per wave
  uint32_t     la = lds_stage + 16u * tid; // mirrored LDS layout
  async_copy16<0 * 4096>(la, ga);
  async_copy16<1 * 4096>(la, ga);
  async_copy16<2 * 4096>(la, ga);
  async_copy16<3 * 4096>(la, ga);
  async_copy16<4 * 4096>(la, ga);
  async_copy16<5 * 4096>(la, ga);
  async_copy16<6 * 4096>(la, ga);
  async_copy16<7 * 4096>(la, ga);
}

// Low 32 bits of a generic pointer to __shared__ == LDS byte offset
// (aperture form {SHARED_BASE, offset}, ISA 10.2).
__device__ __forceinline__ uint32_t lds_addr_of(const void* p) {
  return (uint32_t)(uintptr_t)p;
}

// ---------------- math helpers (single-instruction trans ops) ----------------
__device__ __forceinline__ float softplus_f(float x) {
  // stable: max(x,0) + ln(1 + e^{-|x|});  v_exp_f32/v_log_f32 are base-2.
  const float LOG2E = 1.4426950408889634f;
  const float LN2   = 0.6931471805599453f;
  float e = __builtin_amdgcn_exp2f(-fabsf(x) * LOG2E);   // e^{-|x|}
  float l = __builtin_amdgcn_logf(1.0f + e);             // log2(1+e^{-|x|})
  return fmaxf(x, 0.0f) + LN2 * l;
}

__device__ __forceinline__ float hw_tanh(float x) {
  float r;
  // CDNA5 TRANS op. ISA 7.4: HW does not interlock TRANS result consumption
  // from the same wave -- need 1 independent op or V_NOP before using the
  // output. The compiler guards its own trans ops but not inline asm, so
  // bake the V_NOP into the blob.
  asm("v_tanh_f32 %0, %1\n\tv_nop" : "=v"(r) : "v"(x));
  return r;
}

// ---------------- kernel 1: per-chunk sum of softplus ----------------
__global__ __launch_bounds__(NT) void k_chunk_sums(const float* __restrict__ x,
                                                   float* __restrict__ sums) {
  __shared__ __align__(16) float smem[2 * STAGE_F];
  const int d = threadIdx.x;
  const int c = blockIdx.x;
  const int b = blockIdx.y;
  const uint32_t lds0 = lds_addr_of(smem);
  const float* gchunk = x + ((size_t)b * T_DIM + (size_t)c * CH) * D_DIM;

  issue_stage(gchunk, lds0, d);            // prime buf0
  float s = 0.0f;
  for (int st = 0; st < NSTAGE; ++st) {
    if (st + 1 < NSTAGE) {
      issue_stage(gchunk + (size_t)(st + 1) * STAGE_F,
                  lds0 + (uint32_t)(((st + 1) & 1) * STAGE_BYTES), d);
      WAIT_ASYNC(8);                       // prev stage (8 older ops) complete
    } else {
      WAIT_ASYNC(0);
    }
    __syncthreads();                       // cross-wave visibility of stage st
    const float* sm = smem + (st & 1) * STAGE_F + d;
#pragma unroll 8
    for (int t = 0; t < BUF_T; ++t) s += softplus_f(sm[t * D_DIM]);
    __syncthreads();                       // done reading before buffer reuse
  }
  sums[((size_t)b * NCHUNK + c) * D_DIM + d] = s;
}

// ---------------- kernel 2: exclusive scan of chunk sums + log2(t+1) table ----------------
__global__ __launch_bounds__(NT) void k_scan(float* __restrict__ sums,
                                             float* __restrict__ logc) {
  const int d = threadIdx.x;
  const int b = blockIdx.x;
  float run = 0.0f;
  for (int c = 0; c < NCHUNK; ++c) {
    size_t i = ((size_t)b * NCHUNK + c) * D_DIM + d;
    float v = sums[i];
    sums[i] = run;                         // exclusive prefix (carry-in per chunk)
    run += v;
  }
  int gid = b * NT + d;
  if (gid < T_DIM) logc[gid] = __builtin_amdgcn_logf((float)(gid + 1)); // log2(c)
}

// ---------------- kernel 3: scan with carry + log / tanh-sigmoid ----------------
__global__ __launch_bounds__(NT) void k_final(const float* __restrict__ x,
                                              const float* __restrict__ w,
                                              const float* __restrict__ q,
                                              const float* __restrict__ bias,
                                              const float* __restrict__ sums,
                                              const float* __restrict__ logc,
                                              float* __restrict__ out) {
  __shared__ __align__(16) float smem[2 * STAGE_F];
  const int d = threadIdx.x;
  const int c = blockIdx.x;
  const int b = blockIdx.y;
  const uint32_t lds0 = lds_addr_of(smem);
  const float* gchunk = x + ((size_t)b * T_DIM + (size_t)c * CH) * D_DIM;

  const float LN2  = 0.6931471805599453f;
  const float wln2 = w[d] * LN2;           // y = wln2*log2(s) + (bias - qln2*log2(c))
  const float qln2 = q[0] * LN2;
  const float bv   = bias[0];

  float s = sums[((size_t)b * NCHUNK + c) * D_DIM + d]; // carry-in
  float* ob = out + ((size_t)b * T_DIM + (size_t)c * CH) * D_DIM + d;
  const float* lcb = logc + c * CH;

  issue_stage(gchunk, lds0, d);
  for (int st = 0; st < NSTAGE; ++st) {
    if (st + 1 < NSTAGE) {
      issue_stage(gchunk + (size_t)(st + 1) * STAGE_F,
                  lds0 + (uint32_t)(((st + 1) & 1) * STAGE_BYTES), d);
      WAIT_ASYNC(8);
    } else {
      WAIT_ASYNC(0);
    }
    __syncthreads();
    const float* sm = smem + (st & 1) * STAGE_F + d;
    const float* lc = lcb + st * BUF_T;    // loop-uniform -> scalar-cached loads
    float* op = ob + (size_t)st * STAGE_F;
#pragma unroll 4
    for (int t = 0; t < BUF_T; ++t) {
      s += softplus_f(sm[t * D_DIM]);      // running state (fp32 scan)
      float a = fmaf(-qln2, lc[t], bv);
      float y = fmaf(wln2, __builtin_amdgcn_logf(s), a);
      op[(size_t)t * D_DIM] = fmaf(0.5f, hw_tanh(0.5f * y), 0.5f); // sigmoid
    }
    __syncthreads();
  }
}

// ---------------- launcher ----------------
extern "C" void kernel_launch(void* const* d_in, const int* in_sizes, int n_in,
                              void* d_out, int out_size, void* d_ws, size_t ws_size,
                              hipStream_t stream) {
  (void)in_sizes; (void)n_in; (void)out_size; (void)ws_size;
  const float* x    = (const float*)d_in[0];  // [B,T,D]
  const float* w    = (const float*)d_in[1];  // [D,1]
  const float* q    = (const float*)d_in[2];  // [1,1]
  const float* bias = (const float*)d_in[3];  // [1,1]
  float* out = (float*)d_out;

  float* sums = (float*)d_ws;                              // B*NCHUNK*D = 1 MB
  float* logc = sums + (size_t)B_DIM * NCHUNK * D_DIM;     // +16 KB table

  dim3 grid(NCHUNK, B_DIM);
  k_chunk_sums<<<grid, NT, 0, stream>>>(x, sums);
  k_scan<<<B_DIM, NT, 0, stream>>>(sums, logc);
  k_final<<<grid, NT, 0, stream>>>(x, w, q, bias, sums, logc, out);
}